// TransformerEmbedding_bilinear_57466662420594
// MI455X (gfx1250) — compile-verified
//
#include <hip/hip_runtime.h>
#include <hip/hip_bf16.h>

typedef __attribute__((ext_vector_type(2))) float v2f;
typedef __attribute__((ext_vector_type(8))) float v8f;

#define B_   8
#define S_   512
#define VOCAB_ 32000
#define E_   128
#define D_   512
#define LN_EPS_ 1e-5f

// ---------------------------------------------------------------------------
// Kernel 1: m[b][o][i] = sum_j W[o][i][j] * king[b][j]
// Grid: D_ blocks of E_ threads. Each thread streams one contiguous W row
// (512 B) once and does 8 FMAs per element (one per batch) -> W traffic is
// read-once (33.5 MB), the bandwidth floor for this stage.
// ---------------------------------------------------------------------------
__global__ __launch_bounds__(E_) void compute_m_kernel(
    const float* __restrict__ W,          // [D][E][E]
    const float* __restrict__ king_table, // [N_KINGS][E]
    const int*   __restrict__ king_id,    // [B]
    float*       __restrict__ m)          // [B][D][E]
{
    __shared__ float kings[B_][E_];
    const int o = blockIdx.x;
    const int i = threadIdx.x;

    for (int v = threadIdx.x; v < B_ * E_; v += blockDim.x) {
        int bb = v >> 7, j = v & (E_ - 1);
        kings[bb][j] = king_table[(size_t)king_id[bb] * E_ + j];
    }
    __syncthreads();

    const float* wrow = W + ((size_t)o * E_ + i) * E_;
    float acc[B_];
#pragma unroll
    for (int bb = 0; bb < B_; ++bb) acc[bb] = 0.f;

    for (int j = 0; j < E_; ++j) {
        float w = wrow[j];
#pragma unroll
        for (int bb = 0; bb < B_; ++bb) acc[bb] += w * kings[bb][j];
    }
#pragma unroll
    for (int bb = 0; bb < B_; ++bb)
        m[((size_t)bb * D_ + o) * E_ + i] = acc[bb];
}

// ---------------------------------------------------------------------------
// Kernel 2: x[b][s][o] = sum_i seq_emb[b][s][i] * m[b][o][i] + bias[o],
// then LayerNorm over o. One block = one batch b x 16 sequence rows.
// 8 waves; wave w computes o in [w*64, w*64+64) as 4 16x16 f32 WMMA tiles,
// K=128 via 32 chained V_WMMA_F32_16X16X4_F32.
// ---------------------------------------------------------------------------
__global__ __launch_bounds__(256) void gemm_bias_ln_kernel(
    const int*   __restrict__ sequence,   // [B][S]
    const int*   __restrict__ king_id,    // [B]
    const float* __restrict__ emb_tables, // [N_KINGS][VOCAB][E]
    const float* __restrict__ m,          // [B][D][E]
    const float* __restrict__ bias,       // [D]
    const float* __restrict__ gamma,      // [D]
    const float* __restrict__ beta,       // [D]
    float*       __restrict__ out)        // [B][S][D]
{
    __shared__ float As[16][E_];   // seq_emb tile (8 KB)
    __shared__ float Xs[16][D_];   // result tile  (32 KB)

    const int b   = blockIdx.y;
    const int s0  = blockIdx.x * 16;
    const int tid = threadIdx.x;
    const int king = king_id[b];

    // Gather 16 embedding rows into LDS.
    for (int idx = tid; idx < 16 * E_; idx += 256) {
        int r = idx >> 7, i = idx & (E_ - 1);
        int tok = sequence[b * S_ + s0 + r];
        As[r][i] = emb_tables[((size_t)king * VOCAB_ + tok) * E_ + i];
    }
    __syncthreads();

    const int wave  = tid >> 5;
    const int lane  = tid & 31;
    const int n     = lane & 15;   // A: M row / B: N col within tile
    const int khalf = lane >> 4;   // K-pair select per ISA 16x4 f32 layout
    const int obase = wave * 64;
    const float* mB = m + (size_t)b * D_ * E_;

    v8f acc0 = {}, acc1 = {}, acc2 = {}, acc3 = {};

    for (int k0 = 0; k0 < E_; k0 += 4) {
        const int kk = k0 + 2 * khalf;

        // A fragment: lanes 0-15 hold K={k0,k0+1}, lanes 16-31 K={k0+2,k0+3}
        v2f a;
        a.x = As[n][kk];
        a.y = As[n][kk + 1];

        // B fragments: B[k][n] = m[b][obase + 16t + n][k0 + k]
        const float* p0 = mB + (size_t)(obase +  0 + n) * E_ + kk;
        const float* p1 = mB + (size_t)(obase + 16 + n) * E_ + kk;
        const float* p2 = mB + (size_t)(obase + 32 + n) * E_ + kk;
        const float* p3 = mB + (size_t)(obase + 48 + n) * E_ + kk;
        v2f b0, b1, b2, b3;
        b0.x = p0[0]; b0.y = p0[1];
        b1.x = p1[0]; b1.y = p1[1];
        b2.x = p2[0]; b2.y = p2[1];
        b3.x = p3[0]; b3.y = p3[1];

        acc0 = __builtin_amdgcn_wmma_f32_16x16x4_f32(false, a, false, b0, (short)0, acc0, false, false);
        acc1 = __builtin_amdgcn_wmma_f32_16x16x4_f32(false, a, false, b1, (short)0, acc1, false, false);
        acc2 = __builtin_amdgcn_wmma_f32_16x16x4_f32(false, a, false, b2, (short)0, acc2, false, false);
        acc3 = __builtin_amdgcn_wmma_f32_16x16x4_f32(false, a, false, b3, (short)0, acc3, false, false);
    }

    // C/D layout: VGPR r -> row M = r + 8*khalf, col N = lane&15.
    const int rowoff = khalf * 8;
    const float bias0 = bias[obase +  0 + n];
    const float bias1 = bias[obase + 16 + n];
    const float bias2 = bias[obase + 32 + n];
    const float bias3 = bias[obase + 48 + n];
#pragma unroll
    for (int r = 0; r < 8; ++r) {
        Xs[rowoff + r][obase +  0 + n] = acc0[r] + bias0;
        Xs[rowoff + r][obase + 16 + n] = acc1[r] + bias1;
        Xs[rowoff + r][obase + 32 + n] = acc2[r] + bias2;
        Xs[rowoff + r][obase + 48 + n] = acc3[r] + bias3;
    }
    __syncthreads();

    // LayerNorm: 8 waves x 2 rows each; wave32 shuffle reduction.
#pragma unroll
    for (int rr = 0; rr < 2; ++rr) {
        const int r = wave * 2 + rr;
        float s = 0.f, s2 = 0.f;
        for (int o = lane; o < D_; o += 32) {
            float v = Xs[r][o];
            s  += v;
            s2 += v * v;
        }
#pragma unroll
        for (int off = 16; off > 0; off >>= 1) {
            s  += __shfl_xor(s,  off, 32);
            s2 += __shfl_xor(s2, off, 32);
        }
        const float mean = s * (1.0f / D_);
        const float var  = s2 * (1.0f / D_) - mean * mean;
        const float inv  = rsqrtf(var + LN_EPS_);
        float* orow = out + ((size_t)b * S_ + s0 + r) * D_;
        for (int o = lane; o < D_; o += 32) {
            float v = (Xs[r][o] - mean) * inv;
            orow[o] = v * gamma[o] + beta[o];
        }
    }
}

extern "C" void kernel_launch(void* const* d_in, const int* in_sizes, int n_in,
                              void* d_out, int out_size, void* d_ws, size_t ws_size,
                              hipStream_t stream) {
    const int*   sequence   = (const int*)  d_in[0]; // [B][S]
    const int*   king_id    = (const int*)  d_in[1]; // [B]
    const float* emb_tables = (const float*)d_in[2]; // [27][32000][128]
    const float* king_table = (const float*)d_in[3]; // [27][128]
    const float* W          = (const float*)d_in[4]; // [512][128][128]
    const float* bias       = (const float*)d_in[5]; // [512]
    const float* gamma      = (const float*)d_in[6]; // [512]
    const float* beta       = (const float*)d_in[7]; // [512]
    float*       out        = (float*)d_out;         // [B][S][D]
    float*       m          = (float*)d_ws;          // [B][D][E] = 2 MB scratch

    compute_m_kernel<<<D_, E_, 0, stream>>>(W, king_table, king_id, m);
    gemm_bias_ln_kernel<<<dim3(S_ / 16, B_), 256, 0, stream>>>(
        sequence, king_id, emb_tables, m, bias, gamma, beta, out);
}